// ConvBNReLU_66340064854632
// MI455X (gfx1250) — compile-verified
//
#include <hip/hip_runtime.h>

typedef __attribute__((ext_vector_type(2))) float v2f;
typedef __attribute__((ext_vector_type(8))) float v8f;

#define DFEAT 64
#define BN_EPS 1e-5f

// ---------------- init: deg = 1 (self loop), stats = 0 ----------------
__global__ void gcn_init_kernel(float* __restrict__ deg, float* __restrict__ stats, int N) {
    int i = blockIdx.x * blockDim.x + threadIdx.x;
    if (i < N) deg[i] = 1.0f;
    if (i < 128) stats[i] = 0.0f;
}

// ---------------- degree count over dst ----------------
__global__ void gcn_deg_kernel(const int* __restrict__ dst, float* __restrict__ deg, int E) {
    int e = blockIdx.x * blockDim.x + threadIdx.x;
    if (e < E) atomicAdd(&deg[dst[e]], 1.0f);
}

// ---------------- dinv = rsqrt(deg) ----------------
__global__ void gcn_dinv_kernel(const float* __restrict__ deg, float* __restrict__ dinv, int N) {
    int i = blockIdx.x * blockDim.x + threadIdx.x;
    if (i < N) dinv[i] = rsqrtf(deg[i]);
}

// ---------------- GEMM h = x@W via V_WMMA_F32_16X16X4_F32, fused hs = h*dinv[row]
// One wave per 16x16 C tile. Block = 256 threads = 8 waves = 32 rows x 64 cols.
// A 16x4 f32: lane l -> row l%16, K = 2*(l/16)+{0,1}  (2 VGPRs)
// B 4x16 f32: VGPR v -> K = v + 2*(l/16), N = l%16    (2 VGPRs)
// C 16x16 f32: VGPR v -> row v + 8*(l/16), col l%16   (8 VGPRs)
__global__ void gcn_gemm_wmma_kernel(const float* __restrict__ x, const float* __restrict__ W,
                                     const float* __restrict__ dinv,
                                     float* __restrict__ HS, float* __restrict__ TMP) {
    const int lane = threadIdx.x & 31;
    const int wave = threadIdx.x >> 5;
    const int rowBase = blockIdx.x * 32 + (wave >> 2) * 16;
    const int colBase = (wave & 3) * 16;
    const int lmod = lane & 15;
    const int lhi  = lane >> 4;   // 0 or 1

    v8f c = {};
#pragma unroll
    for (int kk = 0; kk < 16; ++kk) {
        const int k = kk * 4 + 2 * lhi;
        v2f a, b;
        a.x = x[(rowBase + lmod) * DFEAT + k + 0];
        a.y = x[(rowBase + lmod) * DFEAT + k + 1];
        b.x = W[(k + 0) * DFEAT + colBase + lmod];
        b.y = W[(k + 1) * DFEAT + colBase + lmod];
        // 8 args: (neg_a, A, neg_b, B, c_mod, C, reuse_a, reuse_b)
        c = __builtin_amdgcn_wmma_f32_16x16x4_f32(false, a, false, b, (short)0, c,
                                                  false, false);
    }
#pragma unroll
    for (int v = 0; v < 8; ++v) {
        const int row = rowBase + v + 8 * lhi;
        const int col = colBase + lmod;
        const float hs = c[v] * dinv[row];
        HS[row * DFEAT + col]  = hs;  // scaled features for edge gather
        TMP[row * DFEAT + col] = hs;  // accumulator init = self-loop term
    }
}

// ---------------- edge scatter: TMP[dst] += HS[src] (16 threads/edge, float4 each)
__global__ void gcn_scatter_kernel(const float* __restrict__ HS, float* __restrict__ TMP,
                                   const int* __restrict__ src, const int* __restrict__ dst,
                                   int E) {
    int idx = blockIdx.x * blockDim.x + threadIdx.x;
    if (idx >= E * 16) return;
    const int e = idx >> 4;
    const int part = (idx & 15) * 4;
    const int s = src[e];
    const int t = dst[e];
    const float4 v = *reinterpret_cast<const float4*>(HS + s * DFEAT + part);
    float* p = TMP + t * DFEAT + part;
    atomicAdd(p + 0, v.x);
    atomicAdd(p + 1, v.y);
    atomicAdd(p + 2, v.z);
    atomicAdd(p + 3, v.w);
}

// ---------------- agg = dinv[i]*TMP + b, in-place; accumulate BN column stats
// Block = 256 threads: tx = feature (0..63), ty = 0..3; each block does 64 rows.
__global__ void gcn_agg_stats_kernel(float* __restrict__ TMP, const float* __restrict__ dinv,
                                     const float* __restrict__ bias, float* __restrict__ stats) {
    __shared__ float ssum[4][DFEAT];
    __shared__ float ssq[4][DFEAT];
    const int tx = threadIdx.x & 63;
    const int ty = threadIdx.x >> 6;
    const int rowBase = blockIdx.x * 64;
    const float bv = bias[tx];
    float sum = 0.0f, sq = 0.0f;
#pragma unroll 4
    for (int r = ty; r < 64; r += 4) {
        const int i = rowBase + r;
        const float val = dinv[i] * TMP[i * DFEAT + tx] + bv;
        TMP[i * DFEAT + tx] = val;
        sum += val;
        sq += val * val;
    }
    ssum[ty][tx] = sum;
    ssq[ty][tx]  = sq;
    __syncthreads();
    if (ty == 0) {
        const float s = ssum[0][tx] + ssum[1][tx] + ssum[2][tx] + ssum[3][tx];
        const float q = ssq[0][tx] + ssq[1][tx] + ssq[2][tx] + ssq[3][tx];
        atomicAdd(&stats[tx], s);
        atomicAdd(&stats[64 + tx], q);
    }
}

// ---------------- BatchNorm (batch stats) + ReLU, in place on d_out ----------------
__global__ void gcn_bn_relu_kernel(float* __restrict__ out, const float* __restrict__ stats,
                                   const float* __restrict__ gamma, const float* __restrict__ beta,
                                   int N) {
    const int idx = blockIdx.x * blockDim.x + threadIdx.x;
    const int total = N * DFEAT;
    if (idx >= total) return;
    const int d = idx & 63;
    const float invN = 1.0f / (float)N;
    const float mean = stats[d] * invN;
    const float var  = stats[64 + d] * invN - mean * mean;
    const float v = out[idx];
    const float y = gamma[d] * (v - mean) * rsqrtf(var + BN_EPS) + beta[d];
    out[idx] = fmaxf(y, 0.0f);
}

extern "C" void kernel_launch(void* const* d_in, const int* in_sizes, int n_in,
                              void* d_out, int out_size, void* d_ws, size_t ws_size,
                              hipStream_t stream) {
    const float* x     = (const float*)d_in[0];
    const int*   edges = (const int*)d_in[1];   // [2, E]: row0 = src, row1 = dst
    const float* W     = (const float*)d_in[2];
    const float* bias  = (const float*)d_in[3];
    const float* gamma = (const float*)d_in[4];
    const float* beta  = (const float*)d_in[5];

    const int N = in_sizes[0] / DFEAT;   // 65536
    const int E = in_sizes[1] / 2;       // 1048576
    const int* src = edges;
    const int* dst = edges + E;

    float* ws    = (float*)d_ws;
    float* deg   = ws;                 // N
    float* dinv  = ws + N;             // N
    float* stats = ws + 2 * N;         // 128 (sum | sumsq)
    float* HS    = ws + 2 * N + 128;   // N*64, float4-aligned
    float* TMP   = (float*)d_out;      // accumulator lives in d_out

    gcn_init_kernel<<<(N + 255) / 256, 256, 0, stream>>>(deg, stats, N);
    gcn_deg_kernel<<<(E + 255) / 256, 256, 0, stream>>>(dst, deg, E);
    gcn_dinv_kernel<<<(N + 255) / 256, 256, 0, stream>>>(deg, dinv, N);
    gcn_gemm_wmma_kernel<<<N / 32, 256, 0, stream>>>(x, W, dinv, HS, TMP);
    gcn_scatter_kernel<<<(E * 16 + 255) / 256, 256, 0, stream>>>(HS, TMP, src, dst, E);
    gcn_agg_stats_kernel<<<N / 64, 256, 0, stream>>>(TMP, dinv, bias, stats);
    gcn_bn_relu_kernel<<<(N * DFEAT + 255) / 256, 256, 0, stream>>>(TMP, stats, gamma, beta, N);
}